// _AttributeTopicModule_73280732004920
// MI455X (gfx1250) — compile-verified
//
#include <hip/hip_runtime.h>

typedef __attribute__((ext_vector_type(2))) float v2f;
typedef __attribute__((ext_vector_type(8))) float v8f;

#define NBATCH  8
#define MROWS   2048          // C*P rows per batch
#define DIMZ    512
#define DIMG    1024
#define EDIM    128
#define MT      32            // M-rows per workgroup tile
#define QZ_LD   516           // padded LDS strides: conflict-free B64 A-frag reads, 16B rows
#define QG_LD   1028
#define PANEL_KP 16           // W_base k-panel height for GEMM2
#define PANEL_LD 516          // pad so half-wave B-frag rows hit disjoint banks
#define PANEL_SZ (PANEL_KP * PANEL_LD)     // 8256 floats
#define NPANEL   (DIMG / PANEL_KP)         // 64 panels

// float offsets into dynamic LDS
#define QZ_OFF    0
#define QG_OFF    (QZ_OFF + MT * QZ_LD)          // 16512
#define U_OFF     (QG_OFF + MT * QG_LD)          // 49408 (u[1024] ++ v[512] contiguous)
#define V_OFF     (U_OFF + DIMG)                 // 50432
#define ZD_OFF    (V_OFF + DIMZ)                 // 50944
#define QGU_OFF   (ZD_OFF + MT)                  // 50976
#define INV_OFF   (QGU_OFF + MT)                 // 51008
#define PART_OFF  (INV_OFF + MT)                 // 51040
#define PANEL_OFF (PART_OFF + 512)               // 51552 (x4 = 206208 B, 16B aligned)
#define LDS_FLOATS (PANEL_OFF + 2 * PANEL_SZ)    // 68064 floats = 272256 B < 320KB

// ---- CDNA5 async global->LDS copy (ASYNCcnt-tracked), per ISA 15.18.3 ----
__device__ __forceinline__ void async_load_b128_to_lds(unsigned lds_byte_addr,
                                                       const float* gaddr) {
    asm volatile("global_load_async_to_lds_b128 %0, %1, off"
                 :: "v"(lds_byte_addr), "v"(gaddr)
                 : "memory");
}

__device__ __forceinline__ void wait_asynccnt0() {
#if __has_builtin(__builtin_amdgcn_s_wait_asynccnt)
    __builtin_amdgcn_s_wait_asynccnt(0);
#else
    asm volatile("s_wait_asynccnt 0x0" ::: "memory");
#endif
}

// Issue one 16x512 W_base k-panel as coalesced async B128 copies (8 per thread).
__device__ __forceinline__ void issue_panel(const float* __restrict__ W_base,
                                            int p, int tid) {
    const unsigned base = (unsigned)((PANEL_OFF + (p & 1) * PANEL_SZ) * sizeof(float));
    #pragma unroll
    for (int j = 0; j < 8; ++j) {
        const int c   = tid + j * 256;          // chunk 0..2047
        const int row = c >> 7;                 // 0..15
        const int col = (c & 127) * 4;          // float col, 16B chunks
        const float* g = W_base + (size_t)(p * PANEL_KP + row) * DIMZ + col;
        async_load_b128_to_lds(base + (unsigned)((row * PANEL_LD + col) * sizeof(float)), g);
    }
}

// ---- tiny setup kernel: uv[b] = [emb@Wu^T (1024) || emb@Wv^T (512)] ----
__global__ void __launch_bounds__(256, 4)
compute_uv(const int* __restrict__ attr_idx, const float* __restrict__ embed_table,
           const float* __restrict__ Wu, const float* __restrict__ Wv,
           float* __restrict__ uv) {
    __shared__ float emb_s[EDIM];
    const int b = blockIdx.x, tid = threadIdx.x;
    if (tid < EDIM) emb_s[tid] = embed_table[(size_t)attr_idx[b] * EDIM + tid];
    __syncthreads();
    for (int o = tid; o < DIMG + DIMZ; o += 256) {
        const float* wrow = (o < DIMG) ? (Wu + (size_t)o * EDIM)
                                       : (Wv + (size_t)(o - DIMG) * EDIM);
        float s = 0.f;
        #pragma unroll
        for (int e4 = 0; e4 < EDIM / 4; ++e4) {
            float4 w = *(const float4*)(wrow + e4 * 4);
            float4 a = *(const float4*)&emb_s[e4 * 4];
            s += w.x * a.x + w.y * a.y + w.z * a.z + w.w * a.w;
        }
        uv[(size_t)b * (DIMG + DIMZ) + o] = s;
    }
}

// GEMM1 helpers: load 8 B-frags (W_base^T) / issue 16 WMMAs for one k-step
#define LOAD_B1(dst, kk)                                                     \
    {                                                                        \
        _Pragma("unroll") for (int t = 0; t < 8; ++t)                        \
            dst[t] = *(const v2f*)(wb_wave + (size_t)t * (16 * DIMZ) + (kk)); \
    }
#define STEP1(bv, kk)                                                        \
    {                                                                        \
        const v2f a0 = *(const v2f*)&qz_s[(0  + arow) * QZ_LD + (kk) + koff]; \
        const v2f a1 = *(const v2f*)&qz_s[(16 + arow) * QZ_LD + (kk) + koff]; \
        _Pragma("unroll") for (int t = 0; t < 8; ++t) {                      \
            acc[0][t] = __builtin_amdgcn_wmma_f32_16x16x4_f32(               \
                false, a0, false, bv[t], (short)0, acc[0][t], false, false); \
            acc[1][t] = __builtin_amdgcn_wmma_f32_16x16x4_f32(               \
                false, a1, false, bv[t], (short)0, acc[1][t], false, false); \
        }                                                                    \
    }

__global__ void __launch_bounds__(256, 1)
attr_topic_fused(const float* __restrict__ qz,
                 const float* __restrict__ W_base,
                 const float* __restrict__ uv,
                 float*       __restrict__ out)
{
    extern __shared__ float smem[];
    float* qz_s  = smem + QZ_OFF;   // [MT][QZ_LD]
    float* qg_s  = smem + QG_OFF;   // [MT][QG_LD]
    float* u_s   = smem + U_OFF;    // [1024]
    float* v_s   = smem + V_OFF;    // [512]
    float* zd_s  = smem + ZD_OFF;   // [32]
    float* qgu_s = smem + QGU_OFF;  // [32]
    float* inv_s = smem + INV_OFF;  // [32]
    float* part  = smem + PART_OFF; // [2][256]

    const int tid  = threadIdx.x;
    const int lane = tid & 31;
    const int wave = tid >> 5;
    const int b    = blockIdx.x >> 5;       // 8 batches x 32 m-tiles
    const int mt   = blockIdx.x & 31;
    const size_t tile_base = (size_t)b * MROWS * DIMZ + (size_t)mt * MT * DIMZ;

    // Kick off GEMM2's first W_base panel immediately (dedicated LDS buffers).
    issue_panel(W_base, 0, tid);

    // ---- phase 0: load qz tile (32x512) + u||v (1536 floats, coalesced) ----
    const float4* qz4 = (const float4*)(qz + tile_base);
    for (int i = tid; i < MT * (DIMZ / 4); i += 256) {
        const int r  = i >> 7;
        const int c4 = i & 127;
        float4 val = qz4[i];
        *(float4*)&qz_s[r * QZ_LD + c4 * 4] = val;
    }
    {
        const float4* uv4 = (const float4*)(uv + (size_t)b * (DIMG + DIMZ));
        for (int i = tid; i < (DIMG + DIMZ) / 4; i += 256)
            *(float4*)&u_s[i * 4] = uv4[i];
    }
    __syncthreads();

    // ---- phase 1: zdotv[m] = dot(qz[m,:], v) ----
    {
        const int row = tid >> 3, sub = tid & 7;
        float s = 0.f;
        for (int i = 0; i < DIMZ / 8; ++i) {
            const int k = sub * (DIMZ / 8) + i;
            s += qz_s[row * QZ_LD + k] * v_s[k];
        }
        part[tid] = s;
    }
    __syncthreads();
    if (tid < MT) {
        float s = 0.f;
        #pragma unroll
        for (int j = 0; j < 8; ++j) s += part[tid * 8 + j];
        zd_s[tid] = s;
    }
    __syncthreads();

    // ---- phase 2: GEMM1  qg[32x1024] = qz_tile @ W_base^T (K=512, fp32 WMMA) ----
    // wave owns 8 disjoint n-subtiles x BOTH m-subtiles; B-frags ping-pong
    // software-pipelined so global loads overlap the 16-WMMA chain.
    const int arow    = lane & 15;            // A: lane = row (both halves)
    const int koff    = (lane >> 4) * 2;      // A/B: K-pair select per half-wave
    const int bn      = lane & 15;            // B: lane = column
    const int mrow_hi = (lane >> 4) * 8;      // C/D: M = r + 8*(lane>=16)

    v8f acc[2][8];
    #pragma unroll
    for (int ms = 0; ms < 2; ++ms)
        #pragma unroll
        for (int t = 0; t < 8; ++t) acc[ms][t] = v8f{};

    const float* wb_wave = W_base + (size_t)((wave * 8) * 16 + bn) * DIMZ + koff;
    v2f bA[8], bB[8];
    LOAD_B1(bA, 0);
    #pragma unroll 1
    for (int kb = 0; kb < DIMZ; kb += 8) {
        LOAD_B1(bB, kb + 4);
        STEP1(bA, kb);
        const int kn = (kb + 8 < DIMZ) ? kb + 8 : 0;   // clamped prefetch
        LOAD_B1(bA, kn);
        STEP1(bB, kb + 4);
    }

    // ---- GEMM1 epilogue: hoisted LDS scalars, rank-1 + ReLU, spill qg tile,
    //      and fused row-sum / row.u partials via 16-lane shfl_xor reduction ----
    float zval[2][8], uval[8];
    #pragma unroll
    for (int ms = 0; ms < 2; ++ms)
        #pragma unroll
        for (int r = 0; r < 8; ++r) zval[ms][r] = zd_s[ms * 16 + mrow_hi + r];
    #pragma unroll
    for (int t = 0; t < 8; ++t) uval[t] = u_s[(wave * 8 + t) * 16 + bn];

    float rs[2][8], ru[2][8];
    #pragma unroll
    for (int ms = 0; ms < 2; ++ms)
        #pragma unroll
        for (int r = 0; r < 8; ++r) { rs[ms][r] = 0.f; ru[ms][r] = 0.f; }

    #pragma unroll
    for (int ms = 0; ms < 2; ++ms)
        #pragma unroll
        for (int t = 0; t < 8; ++t) {
            const int n = (wave * 8 + t) * 16 + bn;
            #pragma unroll
            for (int r = 0; r < 8; ++r) {
                const int m = ms * 16 + mrow_hi + r;
                float val = acc[ms][t][r] + zval[ms][r] * uval[t];
                val = val > 0.f ? val : 0.f;
                qg_s[m * QG_LD + n] = val;
                rs[ms][r] += val;
                ru[ms][r] += val * uval[t];
            }
        }
    #pragma unroll
    for (int ms = 0; ms < 2; ++ms)
        #pragma unroll
        for (int r = 0; r < 8; ++r) {
            #pragma unroll
            for (int off = 8; off >= 1; off >>= 1) {
                rs[ms][r] += __shfl_xor(rs[ms][r], off, 16);
                ru[ms][r] += __shfl_xor(ru[ms][r], off, 16);
            }
            if ((lane & 15) == 0) {               // one writer per 16-lane half
                const int m = ms * 16 + mrow_hi + r;
                part[m * 8 + wave]       = rs[ms][r];
                part[256 + m * 8 + wave] = ru[ms][r];
            }
        }
    __syncthreads();

    // ---- phase 3: finalize L1 norm factors ----
    if (tid < MT) {
        float s = 0.f, su = 0.f;
        #pragma unroll
        for (int j = 0; j < 8; ++j) {
            s  += part[tid * 8 + j];
            su += part[256 + tid * 8 + j];
        }
        inv_s[tid] = 1.0f / fmaxf(s, 1e-6f);
        qgu_s[tid] = su;
    }

    // ---- phase 4: GEMM2  out[32x512] = (qg/denom) @ W_base (K=1024) + rank-1 ----
    // B from double-buffered async LDS panels; wave owns 4 n-subtiles x 2 m-subtiles.
    v8f acc2[2][4];
    #pragma unroll
    for (int ms = 0; ms < 2; ++ms)
        #pragma unroll
        for (int t = 0; t < 4; ++t) acc2[ms][t] = v8f{};

    #pragma unroll 1
    for (int p = 0; p < NPANEL; ++p) {
        wait_asynccnt0();           // this wave's panel-p chunks landed in LDS
        __syncthreads();            // all waves' chunks landed; everyone left p-1
        if (p + 1 < NPANEL) issue_panel(W_base, p + 1, tid);   // overlaps compute(p)

        const float* pbuf = smem + PANEL_OFF + (p & 1) * PANEL_SZ;
        #pragma unroll
        for (int kk = 0; kk < PANEL_KP / 4; ++kk) {
            const int kb = p * PANEL_KP + kk * 4;   // global K for A frags
            const int kl = kk * 4 + koff;           // local panel row
            const v2f a0 = *(const v2f*)&qg_s[(0  + arow) * QG_LD + kb + koff];
            const v2f a1 = *(const v2f*)&qg_s[(16 + arow) * QG_LD + kb + koff];
            #pragma unroll
            for (int t = 0; t < 4; ++t) {
                const int n = (wave * 4 + t) * 16 + bn;
                v2f bfrag;
                bfrag.x = pbuf[kl * PANEL_LD + n];
                bfrag.y = pbuf[(kl + 1) * PANEL_LD + n];
                acc2[0][t] = __builtin_amdgcn_wmma_f32_16x16x4_f32(
                    false, a0, false, bfrag, (short)0, acc2[0][t], false, false);
                acc2[1][t] = __builtin_amdgcn_wmma_f32_16x16x4_f32(
                    false, a1, false, bfrag, (short)0, acc2[1][t], false, false);
            }
        }
    }

    // ---- GEMM2 epilogue: hoisted LDS scalars, out = inv[m]*(acc + qgu[m]*v[n]) ----
    float invv[2][8], quv[2][8], vv[4];
    #pragma unroll
    for (int ms = 0; ms < 2; ++ms)
        #pragma unroll
        for (int r = 0; r < 8; ++r) {
            const int m = ms * 16 + mrow_hi + r;
            invv[ms][r] = inv_s[m];
            quv[ms][r]  = qgu_s[m];
        }
    #pragma unroll
    for (int t = 0; t < 4; ++t) vv[t] = v_s[(wave * 4 + t) * 16 + bn];

    float* outp = out + tile_base;
    #pragma unroll
    for (int ms = 0; ms < 2; ++ms)
        #pragma unroll
        for (int t = 0; t < 4; ++t) {
            const int n = (wave * 4 + t) * 16 + bn;
            #pragma unroll
            for (int r = 0; r < 8; ++r) {
                const int m = ms * 16 + mrow_hi + r;
                outp[(size_t)m * DIMZ + n] =
                    invv[ms][r] * (acc2[ms][t][r] + quv[ms][r] * vv[t]);
            }
        }
}

extern "C" void kernel_launch(void* const* d_in, const int* in_sizes, int n_in,
                              void* d_out, int out_size, void* d_ws, size_t ws_size,
                              hipStream_t stream) {
    const float* qz          = (const float*)d_in[0];
    const int*   attr_idx    = (const int*)  d_in[1];
    const float* W_base      = (const float*)d_in[2];
    const float* embed_table = (const float*)d_in[3];
    const float* Wu          = (const float*)d_in[4];
    const float* Wv          = (const float*)d_in[5];
    float*       out         = (float*)d_out;
    float*       uv          = (float*)d_ws;   // [8][1536] = 48 KB scratch

    compute_uv<<<dim3(NBATCH), dim3(256), 0, stream>>>(attr_idx, embed_table, Wu, Wv, uv);

    const size_t shmem = (size_t)LDS_FLOATS * sizeof(float);  // ~272 KB < 320 KB/WGP
    (void)hipFuncSetAttribute((const void*)attr_topic_fused,
                              hipFuncAttributeMaxDynamicSharedMemorySize,
                              (int)shmem);
    // 8 batches x 64 m-tiles = 512 workgroups, 8 wave32 waves each
    attr_topic_fused<<<dim3(NBATCH * (MROWS / MT)), dim3(256), shmem, stream>>>(
        qz, W_base, uv, out);
}